// MessageGNN_82712480186689
// MI455X (gfx1250) — compile-verified
//
#include <hip/hip_runtime.h>
#include <hip/hip_bf16.h>

// ---------------------------------------------------------------------------
// MessageGNN on MI455X (gfx1250): fp32 GEMMs via V_WMMA_F32_16X16X4_F32,
// segment means via global_atomic_add_f32, phase-3 scatter fused into node
// update epilogues.
//
// LDS weight layout: K-panels of 4 -> panel[(s*W+col)*4 + r] = W[4s+r][col].
// A lane's B fragment is one 8B-aligned ds_load_b64, bank-conflict-free.
// B fragments for a K-step are batched before the 8 WMMAs so LDS latency
// overlaps matrix ops.  Node kernels split N in half (gridDim.y=2) to get
// 3 blocks/WGP occupancy (LDS ~103KB) for gather-latency hiding.
// ---------------------------------------------------------------------------

#define NV_ 100000
#define NC_ 400000
#define NU_ 64
#define E_  1200000

typedef __attribute__((ext_vector_type(2))) float v2f;
typedef __attribute__((ext_vector_type(8))) float v8f;

#define WMMA4(a, b, c) \
  __builtin_amdgcn_wmma_f32_16x16x4_f32(false, (a), false, (b), (short)0, (c), false, false)

__device__ __forceinline__ float lrelu(float x) { return x >= 0.0f ? x : 0.1f * x; }

// ---------------------------------------------------------------------------
// counts
// ---------------------------------------------------------------------------
__global__ __launch_bounds__(256) void counts_kernel(
    const int* __restrict__ assigns_dst, const int* __restrict__ contains_dst,
    const int* __restrict__ clause_ctx, const int* __restrict__ var_ctx,
    float* __restrict__ cnt_vc, float* __restrict__ cnt_cv,
    float* __restrict__ ctx_cnt_c, float* __restrict__ ctx_cnt_v) {
  int t = blockIdx.x * blockDim.x + threadIdx.x;
  if (t < E_) {
    unsafeAtomicAdd(&cnt_vc[assigns_dst[t]], 1.0f);
    unsafeAtomicAdd(&cnt_cv[contains_dst[t]], 1.0f);
  }
  if (t < NC_) unsafeAtomicAdd(&ctx_cnt_c[clause_ctx[t]], 1.0f);
  if (t < NV_) unsafeAtomicAdd(&ctx_cnt_v[var_ctx[t]], 1.0f);
}

// ---------------------------------------------------------------------------
// edge message GEMM: m = lrelu([edge_feat(4) | emb[src](128)] @ W(132x128) + b)
// scatter-add m into accum[dst].  One wave = one 16-edge tile, all 128 cols.
// ---------------------------------------------------------------------------
#define EK 132  // 33 K-steps of 4

__global__ __launch_bounds__(256) void edge_msg_kernel(
    const float* __restrict__ edge_feat,  // E x 4
    const float* __restrict__ src_emb,    // [*, 128]
    const int* __restrict__ src_idx,      // E
    const int* __restrict__ dst_idx,      // E
    const float* __restrict__ W,          // 132 x 128 row-major
    const float* __restrict__ bias,       // 128
    float* __restrict__ accum,            // [Ndst, 128]
    int num_tiles) {
  __shared__ float Wp[(EK / 4) * 128 * 4 + 128];  // K-panel layout + bias
  float* bl = &Wp[(EK / 4) * 128 * 4];
  for (int t = threadIdx.x; t < EK * 128; t += blockDim.x) {
    int k = t >> 7, n = t & 127;
    Wp[(((k >> 2) << 7 | n) << 2) | (k & 3)] = W[t];
  }
  for (int t = threadIdx.x; t < 128; t += blockDim.x) bl[t] = bias[t];
  __syncthreads();

  const int lane = threadIdx.x & 31;
  const int hi = lane >> 4;
  const int mr = lane & 15;
  const int koff = hi * 2;
  const int wid = blockIdx.x * (blockDim.x >> 5) + (threadIdx.x >> 5);
  const int nwaves = gridDim.x * (blockDim.x >> 5);

  for (int tile = wid; tile < num_tiles; tile += nwaves) {
    const int base = tile * 16;
    const int e = base + mr;
    const int src = src_idx[e];
    const float* __restrict__ arow = src_emb + (size_t)src * 128;
    const float* __restrict__ frow = edge_feat + (size_t)e * 4;

    v8f c[8];
#pragma unroll
    for (int n = 0; n < 8; n++) c[n] = (v8f){0, 0, 0, 0, 0, 0, 0, 0};

    // K-step 0: edge features
    {
      v2f a = {frow[koff], frow[koff + 1]};
      v2f bf[8];
#pragma unroll
      for (int n = 0; n < 8; n++)
        bf[n] = *(const v2f*)&Wp[((n * 16 + mr) << 2) + koff];
#pragma unroll
      for (int n = 0; n < 8; n++) c[n] = WMMA4(a, bf[n], c[n]);
    }
    // K-steps 1..32: source embedding
#pragma unroll 2
    for (int k0 = 4; k0 < EK; k0 += 4) {
      const int q = k0 - 4 + koff;
      v2f a = {arow[q], arow[q + 1]};
      const int s = k0 >> 2;
      v2f bf[8];
#pragma unroll
      for (int n = 0; n < 8; n++)
        bf[n] = *(const v2f*)&Wp[(((s << 7) + n * 16 + mr) << 2) + koff];
#pragma unroll
      for (int n = 0; n < 8; n++) c[n] = WMMA4(a, bf[n], c[n]);
    }
    // epilogue: bias + lrelu + scatter-add
    int didx[8];
#pragma unroll
    for (int j = 0; j < 8; j++) didx[j] = dst_idx[base + j + hi * 8];
#pragma unroll
    for (int n = 0; n < 8; n++) {
      const int col = n * 16 + mr;
      const float bb = bl[col];
#pragma unroll
      for (int j = 0; j < 8; j++) {
        float v = lrelu(c[n][j] + bb);
        unsafeAtomicAdd(&accum[(size_t)didx[j] * 128 + col], v);
      }
    }
  }
}

// ---------------------------------------------------------------------------
// node update GEMM (half of N per block: 64 cols, selected by blockIdx.y):
//   new = lrelu([feats(16) | accum1/cnt1 (128) | seg2 (128) | emb (128)]
//               @ W(400x128) + b)
// seg2_mode 0: seg2 = gather_src[seg2_idx[i]];  mode 1: seg2_accum[i]/cnt2[i]
// Optionally scatter-adds results into scatter_accum[scatter_idx[row]].
// ---------------------------------------------------------------------------
#define NK 400     // 100 K-steps of 4
#define NCOL 64    // columns per block (half of 128)

__global__ __launch_bounds__(256) void node_update_kernel(
    const float* __restrict__ feats,        // N x 16
    const float* __restrict__ accum1,       // N x 128
    const float* __restrict__ cnt1,         // N
    const float* __restrict__ gather_src,   // [*,128]   (mode 0)
    const int* __restrict__ seg2_idx,       // N         (mode 0)
    const float* __restrict__ seg2_accum,   // N x 128   (mode 1)
    const float* __restrict__ cnt2,         // N         (mode 1)
    const float* __restrict__ emb,          // N x 128
    const float* __restrict__ W,            // 400 x 128 row-major
    const float* __restrict__ bias,         // 128
    float* __restrict__ out,                // N x 128
    const int* __restrict__ scatter_idx,    // N or null
    float* __restrict__ scatter_accum,      // 64 x 128
    int num_tiles, int seg2_mode) {
  __shared__ float Wp[(NK / 4) * NCOL * 4 + NCOL];
  float* bl = &Wp[(NK / 4) * NCOL * 4];
  const int col_base = blockIdx.y * NCOL;
  for (int t = threadIdx.x; t < NK * NCOL; t += blockDim.x) {
    int k = t >> 6, cl = t & 63;
    Wp[(((k >> 2) << 6 | cl) << 2) | (k & 3)] = W[k * 128 + col_base + cl];
  }
  for (int t = threadIdx.x; t < NCOL; t += blockDim.x) bl[t] = bias[col_base + t];
  __syncthreads();

  const int lane = threadIdx.x & 31;
  const int hi = lane >> 4;
  const int mr = lane & 15;
  const int koff = hi * 2;
  const int wid = blockIdx.x * (blockDim.x >> 5) + (threadIdx.x >> 5);
  const int nwaves = gridDim.x * (blockDim.x >> 5);

  for (int tile = wid; tile < num_tiles; tile += nwaves) {
    const int base = tile * 16;
    const int i = base + mr;
    const float inv1 = 1.0f / fmaxf(cnt1[i], 1.0f);
    const float* __restrict__ frow = feats + (size_t)i * 16;
    const float* __restrict__ a1row = accum1 + (size_t)i * 128;
    const float* __restrict__ erow = emb + (size_t)i * 128;
    const float* __restrict__ s2row;
    float inv2 = 1.0f;
    if (seg2_mode == 0) {
      s2row = gather_src + (size_t)seg2_idx[i] * 128;
    } else {
      s2row = seg2_accum + (size_t)i * 128;
      inv2 = 1.0f / fmaxf(cnt2[i], 1.0f);
    }

    v8f c[4];
#pragma unroll
    for (int n = 0; n < 4; n++) c[n] = (v8f){0, 0, 0, 0, 0, 0, 0, 0};

    // one WMMA K-step with B panel s: batch 4 B loads, then 4 WMMAs
#define KSTEP(sidx, ax, ay)                                               \
    {                                                                     \
      v2f a_ = {(ax), (ay)};                                              \
      const int s_ = (sidx);                                              \
      v2f bf[4];                                                          \
      _Pragma("unroll")                                                   \
      for (int n = 0; n < 4; n++)                                         \
        bf[n] = *(const v2f*)&Wp[(((s_ << 6) + n * 16 + mr) << 2) + koff];\
      _Pragma("unroll")                                                   \
      for (int n = 0; n < 4; n++) c[n] = WMMA4(a_, bf[n], c[n]);          \
    }

    // segment 1: feats, K = [0,16)
#pragma unroll 2
    for (int k0 = 0; k0 < 16; k0 += 4) {
      const int q = k0 + koff;
      KSTEP(k0 >> 2, frow[q], frow[q + 1]);
    }
    // segment 2: mean1, K = [16,144)
#pragma unroll 2
    for (int k0 = 16; k0 < 144; k0 += 4) {
      const int q = k0 - 16 + koff;
      KSTEP(k0 >> 2, a1row[q] * inv1, a1row[q + 1] * inv1);
    }
    // segment 3: ctx gather or second mean, K = [144,272)
#pragma unroll 2
    for (int k0 = 144; k0 < 272; k0 += 4) {
      const int q = k0 - 144 + koff;
      KSTEP(k0 >> 2, s2row[q] * inv2, s2row[q + 1] * inv2);
    }
    // segment 4: own embedding, K = [272,400)
#pragma unroll 2
    for (int k0 = 272; k0 < 400; k0 += 4) {
      const int q = k0 - 272 + koff;
      KSTEP(k0 >> 2, erow[q], erow[q + 1]);
    }
#undef KSTEP

    // epilogue: bias + lrelu + store + optional ctx scatter
    int sidx[8];
    if (scatter_idx) {
#pragma unroll
      for (int j = 0; j < 8; j++) sidx[j] = scatter_idx[base + j + hi * 8];
    }
#pragma unroll
    for (int n = 0; n < 4; n++) {
      const int cl = n * 16 + mr;
      const int col = col_base + cl;
      const float bb = bl[cl];
#pragma unroll
      for (int j = 0; j < 8; j++) {
        float v = lrelu(c[n][j] + bb);
        const int row = base + j + hi * 8;
        out[(size_t)row * 128 + col] = v;
        if (scatter_idx) {
          unsafeAtomicAdd(&scatter_accum[(size_t)sidx[j] * 128 + col], v);
        }
      }
    }
  }
}

// ---------------------------------------------------------------------------
// launch
// ---------------------------------------------------------------------------
extern "C" void kernel_launch(void* const* d_in, const int* in_sizes, int n_in,
                              void* d_out, int out_size, void* d_ws, size_t ws_size,
                              hipStream_t stream) {
  const float* var_emb      = (const float*)d_in[0];
  const float* clause_emb   = (const float*)d_in[1];
  const float* ctx_emb      = (const float*)d_in[2];
  const float* var_feats    = (const float*)d_in[3];
  const float* clause_feats = (const float*)d_in[4];
  const float* ctx_feats    = (const float*)d_in[5];
  const float* edge_sat_vc  = (const float*)d_in[6];
  const float* edge_sat_cv  = (const float*)d_in[7];
  const float* W_vc = (const float*)d_in[8];
  const float* b_vc = (const float*)d_in[9];
  const float* W_cv = (const float*)d_in[10];
  const float* b_cv = (const float*)d_in[11];
  const float* W_c  = (const float*)d_in[12];
  const float* b_c  = (const float*)d_in[13];
  const float* W_v  = (const float*)d_in[14];
  const float* b_v  = (const float*)d_in[15];
  const float* W_u  = (const float*)d_in[16];
  const float* b_u  = (const float*)d_in[17];
  const int* assigns_src  = (const int*)d_in[18];
  const int* assigns_dst  = (const int*)d_in[19];
  const int* contains_src = (const int*)d_in[20];
  const int* contains_dst = (const int*)d_in[21];
  const int* var_ctx      = (const int*)d_in[22];
  const int* clause_ctx   = (const int*)d_in[23];

  float* out = (float*)d_out;

  // workspace layout (floats)
  float* ws = (float*)d_ws;
  float* accum_vc    = ws;                           // NC x 128
  float* accum_cv    = accum_vc + (size_t)NC_ * 128; // NV x 128
  float* cnt_vc      = accum_cv + (size_t)NV_ * 128; // NC
  float* cnt_cv      = cnt_vc + NC_;                 // NV
  float* ctx_accum_c = cnt_cv + NV_;                 // 64 x 128
  float* ctx_accum_v = ctx_accum_c + NU_ * 128;      // 64 x 128
  float* ctx_cnt_c   = ctx_accum_v + NU_ * 128;      // 64
  float* ctx_cnt_v   = ctx_cnt_c + NU_;              // 64
  const size_t ws_floats =
      (size_t)NC_ * 128 + (size_t)NV_ * 128 + NC_ + NV_ + 2 * NU_ * 128 + 2 * NU_;
  hipMemsetAsync(d_ws, 0, ws_floats * sizeof(float), stream);

  // 1) counts
  counts_kernel<<<(E_ + 255) / 256, 256, 0, stream>>>(
      assigns_dst, contains_dst, clause_ctx, var_ctx,
      cnt_vc, cnt_cv, ctx_cnt_c, ctx_cnt_v);

  // 2) edge messages: var -> clause, clause -> var
  edge_msg_kernel<<<1024, 256, 0, stream>>>(
      edge_sat_vc, var_emb, assigns_src, assigns_dst, W_vc, b_vc,
      accum_vc, E_ / 16);
  edge_msg_kernel<<<1024, 256, 0, stream>>>(
      edge_sat_cv, clause_emb, contains_src, contains_dst, W_cv, b_cv,
      accum_cv, E_ / 16);

  // 3) node updates (fused phase-3 scatter into ctx accumulators)
  node_update_kernel<<<dim3(512, 2), 256, 0, stream>>>(
      clause_feats, accum_vc, cnt_vc,
      ctx_emb, clause_ctx, nullptr, nullptr,
      clause_emb, W_c, b_c,
      out, clause_ctx, ctx_accum_c, NC_ / 16, /*seg2_mode=*/0);
  node_update_kernel<<<dim3(512, 2), 256, 0, stream>>>(
      var_feats, accum_cv, cnt_cv,
      ctx_emb, var_ctx, nullptr, nullptr,
      var_emb, W_v, b_v,
      out + (size_t)NC_ * 128, var_ctx, ctx_accum_v, NV_ / 16, /*seg2_mode=*/0);

  // 4) context update (64 rows -> 4 tiles, two small blocks)
  node_update_kernel<<<dim3(1, 2), 128, 0, stream>>>(
      ctx_feats, ctx_accum_c, ctx_cnt_c,
      nullptr, nullptr, ctx_accum_v, ctx_cnt_v,
      ctx_emb, W_u, b_u,
      out + ((size_t)NC_ + NV_) * 128, nullptr, nullptr, NU_ / 16, /*seg2_mode=*/1);
}